// Model_62886911148434
// MI455X (gfx1250) — compile-verified
//
#include <hip/hip_runtime.h>
#include <float.h>
#include <math.h>

// Problem constants (match reference)
#define BB 2048   // sentences
#define LL 64     // maxlen
#define SS 2048   // seeds
#define CC 64     // clusters
#define LS 8      // maxlen_seed
#define DD 300    // embedding dim
#define DP 304    // padded K (multiple of 4, keeps float2 loads 8B-aligned: 304*4=1216)
#define ANPHA 0.5f

typedef __attribute__((ext_vector_type(2))) float v2f;
typedef __attribute__((ext_vector_type(8))) float v8f;

// ---------------------------------------------------------------------------
// CAS-based float atomic max (safe on any arch; values may be negative)
// ---------------------------------------------------------------------------
__device__ __forceinline__ void atomic_max_f32(float* addr, float val) {
    unsigned int* ua = (unsigned int*)addr;
    unsigned int old = *ua;
    while (__uint_as_float(old) < val) {
        unsigned int assumed = old;
        old = atomicCAS(ua, assumed, __float_as_uint(val));
        if (old == assumed) break;
    }
}

// ---------------------------------------------------------------------------
// Kernel 1: init.  Block 0 also builds the cumsum + per-column segment table.
//   pmax_t / pmax_s [B*C] <- -FLT_MAX ; out[0] <- 0 ; seg[s] via searchsorted
// ---------------------------------------------------------------------------
__global__ void init_kernel(const int* __restrict__ num_clusters,
                            float* __restrict__ pmax_t,
                            float* __restrict__ pmax_s,
                            int* __restrict__ seg,
                            float* __restrict__ out) {
    int gid = blockIdx.x * blockDim.x + threadIdx.x;
    // spread -inf init of both [B,C] buffers over the whole grid
    if (gid < BB * CC) pmax_t[gid] = -FLT_MAX;
    if (gid < BB * CC) pmax_s[gid] = -FLT_MAX;

    if (blockIdx.x == 0) {
        __shared__ int csum[CC];
        if (threadIdx.x == 0) {
            out[0] = 0.0f;
            int acc = 0;
            for (int c = 0; c < CC; ++c) { acc += num_clusters[c]; csum[c] = acc; }
        }
        __syncthreads();
        // seg[s] = first c with csum[c] > s   (searchsorted right)
        for (int s = threadIdx.x; s < SS; s += blockDim.x) {
            int c = 0;
            while (c < CC - 1 && csum[c] <= s) ++c;
            seg[s] = c;
        }
    }
}

// ---------------------------------------------------------------------------
// Kernel 2: gather + pooling for both branches.
//   blocks [0,B)   : masked-mean sentence embedding -> snt_t/snt_s [B, DP]
//   blocks [B,B+S) : mean seed-phrase embedding     -> sd_t/sd_s  [S, DP]
// blockDim = 320: one thread per dim (300 active), coalesced row gathers.
// Embedding tables (120 MB total) are L2-resident on MI455X (192 MB L2).
// ---------------------------------------------------------------------------
__global__ void pool_kernel(const int* __restrict__ sents,
                            const int* __restrict__ seeds,
                            const float* __restrict__ mask,
                            const float* __restrict__ emb_t,
                            const float* __restrict__ emb_s,
                            float* __restrict__ snt_t,
                            float* __restrict__ snt_s,
                            float* __restrict__ sd_t,
                            float* __restrict__ sd_s) {
    int bid = blockIdx.x;
    int d = threadIdx.x;
    if (bid < BB) {
        float accT = 0.f, accS = 0.f, msum = 0.f;
        for (int t = 0; t < LL; ++t) {
            int tok = sents[bid * LL + t];
            float mk = mask[bid * LL + t];
            msum += mk;
            if (d < DD) {
                accT += mk * emb_t[(long)tok * DD + d];
                accS += mk * emb_s[(long)tok * DD + d];
            }
        }
        float denom = fmaxf(msum, 1.0f);
        if (d < DP) {
            snt_t[bid * DP + d] = (d < DD) ? accT / denom : 0.0f;
            snt_s[bid * DP + d] = (d < DD) ? accS / denom : 0.0f;
        }
    } else {
        int sid = bid - BB;
        float accT = 0.f, accS = 0.f;
        for (int t = 0; t < LS; ++t) {
            int tok = seeds[sid * LS + t];
            if (d < DD) {
                accT += emb_t[(long)tok * DD + d];
                accS += emb_s[(long)tok * DD + d];
            }
        }
        if (d < DP) {
            sd_t[sid * DP + d] = (d < DD) ? accT * (1.0f / LS) : 0.0f;
            sd_s[sid * DP + d] = (d < DD) ? accS * (1.0f / LS) : 0.0f;
        }
    }
}

// ---------------------------------------------------------------------------
// Kernel 3: pro = snt @ sd^T via V_WMMA_F32_16X16X4_F32, fused with ragged
// per-cluster atomic max.  One wave per 16(M) x 64(N) strip; the A fragment
// is reused across 4 B fragments (4 WMMAs / K-step, 76 K-steps).
//
// f32 WMMA operand mapping (ISA 7.12.2):
//   A (16x4):  lane l -> row l&15,  K pair (l>>4)*2      (v2f)
//   B (4x16):  lane l -> col l&15,  K pair (l>>4)*2      (v2f)  [B=sd row-major]
//   C (16x16): VGPR j, lane l -> M = j + 8*(l>>4), N = l&15
// ---------------------------------------------------------------------------
__global__ void __launch_bounds__(32)
gemm_segmax_kernel(const float* __restrict__ snt,   // [B, DP]
                   const float* __restrict__ sd,    // [S, DP]
                   const int* __restrict__ seg,     // [S]
                   float* __restrict__ pmax) {      // [B, C]
    const int lane = threadIdx.x;            // 0..31, wave32
    const int mn   = lane & 15;
    const int koff = (lane >> 4) * 2;
    const int row0 = blockIdx.y * 16;
    const int col0 = blockIdx.x * 64;

    const float* arow = snt + (long)(row0 + mn) * DP + koff;
    const float* brow0 = sd + (long)(col0 + mn) * DP + koff;
    const float* brow1 = brow0 + 16 * DP;
    const float* brow2 = brow0 + 32 * DP;
    const float* brow3 = brow0 + 48 * DP;

    v8f acc0 = {}, acc1 = {}, acc2 = {}, acc3 = {};

    for (int k = 0; k < DP; k += 4) {
        __builtin_prefetch(arow + k + 32, 0, 3);   // global_prefetch_b8
        v2f a  = *(const v2f*)(arow + k);
        v2f b0 = *(const v2f*)(brow0 + k);
        v2f b1 = *(const v2f*)(brow1 + k);
        v2f b2 = *(const v2f*)(brow2 + k);
        v2f b3 = *(const v2f*)(brow3 + k);
        acc0 = __builtin_amdgcn_wmma_f32_16x16x4_f32(false, a, false, b0,
                                                     (short)0, acc0, false, false);
        acc1 = __builtin_amdgcn_wmma_f32_16x16x4_f32(false, a, false, b1,
                                                     (short)0, acc1, false, false);
        acc2 = __builtin_amdgcn_wmma_f32_16x16x4_f32(false, a, false, b2,
                                                     (short)0, acc2, false, false);
        acc3 = __builtin_amdgcn_wmma_f32_16x16x4_f32(false, a, false, b3,
                                                     (short)0, acc3, false, false);
    }

    // Fused ragged segment max: column -> cluster id, atomic max into [B,C].
    const int mrow = row0 + 8 * (lane >> 4);
    v8f accs[4] = {acc0, acc1, acc2, acc3};
#pragma unroll
    for (int j = 0; j < 4; ++j) {
        int c = col0 + 16 * j + mn;
        int sg = seg[c];
#pragma unroll
        for (int i = 0; i < 8; ++i) {
            atomic_max_f32(&pmax[(long)(mrow + i) * CC + sg], accs[j][i]);
        }
    }
}

// ---------------------------------------------------------------------------
// Kernel 4: per-row softmax(teacher max) + reliability + weighted SE vs
// student max; atomicAdd of the row partial into the scalar loss.
// One 64-thread block per row (one thread per cluster).
// ---------------------------------------------------------------------------
__global__ void loss_kernel(const float* __restrict__ pmax_t,
                            const float* __restrict__ pmax_s,
                            float* __restrict__ out) {
    __shared__ float red[CC];
    int b = blockIdx.x, c = threadIdx.x;
    float t = pmax_t[b * CC + c];

    red[c] = t; __syncthreads();
    for (int off = CC / 2; off > 0; off >>= 1) {
        if (c < off) red[c] = fmaxf(red[c], red[c + off]);
        __syncthreads();
    }
    float m = red[0]; __syncthreads();

    float e = __expf(t - m);
    red[c] = e; __syncthreads();
    for (int off = CC / 2; off > 0; off >>= 1) {
        if (c < off) red[c] += red[c + off];
        __syncthreads();
    }
    float sum = red[0]; __syncthreads();

    float p = e / sum;
    // max softmax prob corresponds to max logit: exp(0)/sum
    float reli = 1.0f / sum - 1.0f / (float)CC;
    float d = p - pmax_s[b * CC + c];
    float w = (1.0f + ANPHA * fabsf(reli)) * d * d;

    red[c] = w; __syncthreads();
    for (int off = CC / 2; off > 0; off >>= 1) {
        if (c < off) red[c] += red[c + off];
        __syncthreads();
    }
    if (c == 0) atomicAdd(out, red[0] * (1.0f / (float)BB));
}

// ---------------------------------------------------------------------------
// Launch
// ---------------------------------------------------------------------------
extern "C" void kernel_launch(void* const* d_in, const int* in_sizes, int n_in,
                              void* d_out, int out_size, void* d_ws, size_t ws_size,
                              hipStream_t stream) {
    const int*   sents        = (const int*)d_in[0];
    const int*   seeds        = (const int*)d_in[1];
    const int*   num_clusters = (const int*)d_in[2];   // [1, C]
    // d_in[3] = num_arr (unused), d_in[5] = flag (static 1 -> loss path)
    const float* mask         = (const float*)d_in[4];
    const float* emb_t        = (const float*)d_in[6];
    const float* emb_s        = (const float*)d_in[7];
    float* out = (float*)d_out;

    // workspace layout (floats): ~11 MB total
    float* snt_t  = (float*)d_ws;               // B*DP
    float* snt_s  = snt_t + (size_t)BB * DP;
    float* sd_t   = snt_s + (size_t)BB * DP;
    float* sd_s   = sd_t + (size_t)SS * DP;
    float* pmax_t = sd_s + (size_t)SS * DP;     // B*C
    float* pmax_s = pmax_t + (size_t)BB * CC;
    int*   seg    = (int*)(pmax_s + (size_t)BB * CC);   // S ints

    // 1) init (-inf maxes, segment table, zero loss)
    init_kernel<<<(BB * CC + 255) / 256, 256, 0, stream>>>(
        num_clusters, pmax_t, pmax_s, seg, out);

    // 2) gather + pooling for sentences and seeds, teacher + student
    pool_kernel<<<BB + SS, 320, 0, stream>>>(
        sents, seeds, mask, emb_t, emb_s, snt_t, snt_s, sd_t, sd_s);

    // 3) f32 WMMA GEMM + fused ragged segment max (teacher, student)
    dim3 ggrid(SS / 64, BB / 16);
    gemm_segmax_kernel<<<ggrid, 32, 0, stream>>>(snt_t, sd_t, seg, pmax_t);
    gemm_segmax_kernel<<<ggrid, 32, 0, stream>>>(snt_s, sd_s, seg, pmax_s);

    // 4) softmax + reliability-weighted squared error -> scalar loss
    loss_kernel<<<BB, CC, 0, stream>>>(pmax_t, pmax_s, out);
}